// CU_24326694764601
// MI455X (gfx1250) — compile-verified
//
#include <hip/hip_runtime.h>

typedef float v2f __attribute__((ext_vector_type(2)));
typedef float v8f __attribute__((ext_vector_type(8)));

#define PLN (1024*1024)          // one 1024x1024 float plane
#define NTOT (1 << 24)           // state size
#define LDA_S 20                 // A_lds row stride (floats): 16 + 4 pad, 16B aligned
#define LDB_S 72                 // B_lds row stride (floats): 64 + 8 pad, 16B aligned

// ---------------------------------------------------------------------------
// Build X = (A - A^H) / 2^9  and  B0 = X/12 + I   (Horner seed, 12-term Taylor)
// Layout of complex matrices in ws: [k][re|im][1024*1024]
// ---------------------------------------------------------------------------
__global__ __launch_bounds__(256)
void CU_skew_seed(const float* __restrict__ Ur, const float* __restrict__ Ui,
                  float* __restrict__ X, float* __restrict__ B0)
{
    long idx = (long)blockIdx.x * blockDim.x + threadIdx.x;   // over 2*PLN
    if (idx >= 2L * PLN) return;
    int k  = (int)(idx >> 20);
    int ij = (int)(idx & (PLN - 1));
    int i = ij >> 10, j = ij & 1023;
    long kij = (long)k * PLN + ij;
    long kji = (long)k * PLN + (long)j * 1024 + i;
    const float sc = 1.0f / 512.0f;                 // 2^-9 scaling
    float xr = (Ur[kij] - Ur[kji]) * sc;            // re(A - A^H)
    float xi = (Ui[kij] + Ui[kji]) * sc;            // im(A - A^H)
    float* Xre = X  + (long)k * 2 * PLN;  float* Xim = Xre + PLN;
    float* Bre = B0 + (long)k * 2 * PLN;  float* Bim = Bre + PLN;
    Xre[ij] = xr;  Xim[ij] = xi;
    float d = (i == j) ? 1.0f : 0.0f;
    Bre[ij] = xr * (1.0f / 12.0f) + d;
    Bim[ij] = xi * (1.0f / 12.0f);
}

// ---------------------------------------------------------------------------
// Complex GEMM core (per 32x32 wave tile), shared by expm and the contraction.
// Consumes one K-chunk of 16 from LDS; 4 WMMAs per (mi,ni,kstep).
// ---------------------------------------------------------------------------
__device__ __forceinline__
void cwmma_chunk(const float* AsRe, const float* AsIm,
                 const float* BsRe, const float* BsIm,
                 int wm, int wn, int l15, int khalf,
                 v8f accRe[2][2], v8f accIm[2][2])
{
#pragma unroll
    for (int ks = 0; ks < 16; ks += 4) {
        int kk = ks + khalf;                         // per-lane K (ISA A/B layout)
        v2f aRe[2], aIm[2], aImN[2], bRe[2], bIm[2];
#pragma unroll
        for (int mi = 0; mi < 2; ++mi) {
            int m = wm + mi * 16 + l15;
            aRe[mi]  = v2f{AsRe[m * LDA_S + kk], AsRe[m * LDA_S + kk + 1]};
            aIm[mi]  = v2f{AsIm[m * LDA_S + kk], AsIm[m * LDA_S + kk + 1]};
            aImN[mi] = -aIm[mi];                     // f32 WMMA has no A-neg modifier
        }
#pragma unroll
        for (int ni = 0; ni < 2; ++ni) {
            int n = wn + ni * 16 + l15;
            bRe[ni] = v2f{BsRe[kk * LDB_S + n], BsRe[(kk + 1) * LDB_S + n]};
            bIm[ni] = v2f{BsIm[kk * LDB_S + n], BsIm[(kk + 1) * LDB_S + n]};
        }
#pragma unroll
        for (int mi = 0; mi < 2; ++mi)
#pragma unroll
            for (int ni = 0; ni < 2; ++ni) {
                accRe[mi][ni] = __builtin_amdgcn_wmma_f32_16x16x4_f32(
                    false, aRe[mi],  false, bRe[ni], (short)0, accRe[mi][ni], false, false);
                accRe[mi][ni] = __builtin_amdgcn_wmma_f32_16x16x4_f32(
                    false, aImN[mi], false, bIm[ni], (short)0, accRe[mi][ni], false, false);
                accIm[mi][ni] = __builtin_amdgcn_wmma_f32_16x16x4_f32(
                    false, aRe[mi],  false, bIm[ni], (short)0, accIm[mi][ni], false, false);
                accIm[mi][ni] = __builtin_amdgcn_wmma_f32_16x16x4_f32(
                    false, aIm[mi],  false, bRe[ni], (short)0, accIm[mi][ni], false, false);
            }
    }
}

// ---------------------------------------------------------------------------
// C = alpha*(A@B) (+ I)   for 1024x1024 complex matrices, grid.z = control k.
// Used for Horner steps (alpha=1/j, addI=1) and squarings (alpha=1, addI=0).
// ---------------------------------------------------------------------------
__global__ __launch_bounds__(128)
void CU_cgemm1024(const float* __restrict__ A, const float* __restrict__ B,
                  float* __restrict__ C, float alpha, int addI)
{
    const float* Are = A + (long)blockIdx.z * 2 * PLN; const float* Aim = Are + PLN;
    const float* Bre = B + (long)blockIdx.z * 2 * PLN; const float* Bim = Bre + PLN;
    float*       Cre = C + (long)blockIdx.z * 2 * PLN; float*       Cim = Cre + PLN;
    int m0 = blockIdx.y * 64, n0 = blockIdx.x * 64;

    __shared__ float AsRe[64 * LDA_S], AsIm[64 * LDA_S];
    __shared__ float BsRe[16 * LDB_S], BsIm[16 * LDB_S];

    int tid = threadIdx.x, lane = tid & 31, wave = tid >> 5;
    int wm = (wave >> 1) * 32, wn = (wave & 1) * 32;
    int l15 = lane & 15, khalf = (lane >> 4) * 2;

    v8f accRe[2][2] = {}; v8f accIm[2][2] = {};

    for (int kc = 0; kc < 1024; kc += 16) {
        for (int i = tid; i < 256; i += 128) {       // A tile 64x16 (2 float4/thread)
            int r = i >> 2, c4 = (i & 3) << 2;
            float4 vr = *(const float4*)(Are + (long)(m0 + r) * 1024 + kc + c4);
            float4 vi = *(const float4*)(Aim + (long)(m0 + r) * 1024 + kc + c4);
            *(float4*)(AsRe + r * LDA_S + c4) = vr;
            *(float4*)(AsIm + r * LDA_S + c4) = vi;
        }
        for (int i = tid; i < 256; i += 128) {       // B tile 16x64
            int r = i >> 4, c4 = (i & 15) << 2;
            float4 vr = *(const float4*)(Bre + (long)(kc + r) * 1024 + n0 + c4);
            float4 vi = *(const float4*)(Bim + (long)(kc + r) * 1024 + n0 + c4);
            *(float4*)(BsRe + r * LDB_S + c4) = vr;
            *(float4*)(BsIm + r * LDB_S + c4) = vi;
        }
        __syncthreads();
        cwmma_chunk(AsRe, AsIm, BsRe, BsIm, wm, wn, l15, khalf, accRe, accIm);
        __syncthreads();
    }

    int rr0 = (lane >> 4) * 8;                       // C/D layout: M = v + 8*(lane>>4)
#pragma unroll
    for (int mi = 0; mi < 2; ++mi)
#pragma unroll
        for (int ni = 0; ni < 2; ++ni) {
            int col = n0 + wn + ni * 16 + l15;
#pragma unroll
            for (int v = 0; v < 8; ++v) {
                int row = m0 + wm + mi * 16 + rr0 + v;
                float cr = accRe[mi][ni][v] * alpha;
                float ci = accIm[mi][ni][v] * alpha;
                if (addI && row == col) cr += 1.0f;
                Cre[(long)row * 1024 + col] = cr;
                Cim[(long)row * 1024 + col] = ci;
            }
        }
}

// ---------------------------------------------------------------------------
// out[k,t,r] = sum_d E[k,t,d] * x[k<<23 | (r>>12)<<22 | d<<12 | (r&4095)]
// M=1024, K=1024, N=8192 per control k. The permutation is pure index math:
// B columns are contiguous runs of 4096 along r, stride 4096 along d.
// ---------------------------------------------------------------------------
__global__ __launch_bounds__(128)
void CU_apply(const float* __restrict__ E,
              const float* __restrict__ xre, const float* __restrict__ xim,
              float* __restrict__ out)
{
    const float* Are = E + (long)blockIdx.z * 2 * PLN; const float* Aim = Are + PLN;
    int m0 = blockIdx.y * 64;                        // t tile
    int n0 = blockIdx.x * 64;                        // r tile (0..8191), 64-aligned
    long kbase = ((long)blockIdx.z << 23) | ((long)((n0 >> 12) & 1) << 22);
    int rlo0 = n0 & 4095;                            // whole tile inside one 4096 run

    __shared__ float AsRe[64 * LDA_S], AsIm[64 * LDA_S];
    __shared__ float BsRe[16 * LDB_S], BsIm[16 * LDB_S];

    int tid = threadIdx.x, lane = tid & 31, wave = tid >> 5;
    int wm = (wave >> 1) * 32, wn = (wave & 1) * 32;
    int l15 = lane & 15, khalf = (lane >> 4) * 2;

    v8f accRe[2][2] = {}; v8f accIm[2][2] = {};

    for (int kc = 0; kc < 1024; kc += 16) {
        for (int i = tid; i < 256; i += 128) {       // E tile 64x16
            int r = i >> 2, c4 = (i & 3) << 2;
            float4 vr = *(const float4*)(Are + (long)(m0 + r) * 1024 + kc + c4);
            float4 vi = *(const float4*)(Aim + (long)(m0 + r) * 1024 + kc + c4);
            *(float4*)(AsRe + r * LDA_S + c4) = vr;
            *(float4*)(AsIm + r * LDA_S + c4) = vi;
        }
        for (int i = tid; i < 256; i += 128) {       // x tile 16x64, bit-mapped gather
            int r = i >> 4, c4 = (i & 15) << 2;
            long g = kbase + ((long)(kc + r) << 12) + rlo0 + c4;
            float4 vr = *(const float4*)(xre + g);
            float4 vi = *(const float4*)(xim + g);
            *(float4*)(BsRe + r * LDB_S + c4) = vr;
            *(float4*)(BsIm + r * LDB_S + c4) = vi;
        }
        __syncthreads();
        cwmma_chunk(AsRe, AsIm, BsRe, BsIm, wm, wn, l15, khalf, accRe, accIm);
        __syncthreads();
    }

    int rr0 = (lane >> 4) * 8;
#pragma unroll
    for (int mi = 0; mi < 2; ++mi)
#pragma unroll
        for (int ni = 0; ni < 2; ++ni) {
            int cc = wn + ni * 16 + l15;             // column within tile (0..63)
#pragma unroll
            for (int v = 0; v < 8; ++v) {
                int t = m0 + wm + mi * 16 + rr0 + v; // target row 0..1023
                long n = kbase + ((long)t << 12) + rlo0 + cc;
                out[n]        = accRe[mi][ni][v];    // real plane
                out[NTOT + n] = accIm[mi][ni][v];    // imag plane
            }
        }
}

// ---------------------------------------------------------------------------
extern "C" void kernel_launch(void* const* d_in, const int* in_sizes, int n_in,
                              void* d_out, int out_size, void* d_ws, size_t ws_size,
                              hipStream_t stream) {
    const float* xre = (const float*)d_in[0];
    const float* xim = (const float*)d_in[1];
    const float* Ur  = (const float*)d_in[2];
    const float* Ui  = (const float*)d_in[3];
    float* out = (float*)d_out;
    float* ws  = (float*)d_ws;

    float* X  = ws;                 // [k][re|im][PLN]  scaled skew generator
    float* B0 = ws + 4L * PLN;      // Horner / squaring ping
    float* B1 = ws + 8L * PLN;      // Horner / squaring pong   (48 MB total)

    CU_skew_seed<<<(2 * PLN + 255) / 256, 256, 0, stream>>>(Ur, Ui, X, B0);

    dim3 blk(128);
    dim3 gG(1024 / 64, 1024 / 64, 2);
    float* cur = B0; float* nxt = B1;
    for (int j = 11; j >= 1; --j) {                  // Horner: B <- (X*B)/j + I
        CU_cgemm1024<<<gG, blk, 0, stream>>>(X, cur, nxt, 1.0f / (float)j, 1);
        float* t = cur; cur = nxt; nxt = t;
    }
    for (int s = 0; s < 9; ++s) {                    // undo 2^-9 scaling: square 9x
        CU_cgemm1024<<<gG, blk, 0, stream>>>(cur, cur, nxt, 1.0f, 0);
        float* t = cur; cur = nxt; nxt = t;
    }

    dim3 gA(8192 / 64, 1024 / 64, 2);                // state contraction
    CU_apply<<<gA, blk, 0, stream>>>(cur, xre, xim, out);
}